// VQCLayer_137438954292
// MI455X (gfx1250) — compile-verified
//
#include <hip/hip_runtime.h>
#include <math.h>

typedef float v2f __attribute__((ext_vector_type(2)));
typedef float v8f __attribute__((ext_vector_type(8)));

#define NQ 10
#define NL 4
#define DIM 1024          // 2^10 statevector
#define BATCH 4096
#define K_IN 128
#define N_OUT 256
#define BPAD 132          // padded LDS row stride (dwords) to spread banks

// ---------------------------------------------------------------------------
// Kernel 1: h = tanh(x @ W_in^T + b_in)   (4096x128)(128x10) -> (4096,10)
// W_in^T is staged once per block into a zero-padded 16x128 LDS tile, so the
// K-loop is branch-free: global_load_b64 (A) + ds_load_b64 (B) + wmma.
// ---------------------------------------------------------------------------
__global__ __launch_bounds__(128) void k_encode(
    const float* __restrict__ x, const float* __restrict__ Win,
    const float* __restrict__ bin, float* __restrict__ h)
{
  __shared__ float Bs[16 * BPAD];   // rows = N (padded cols of W_in), 8.25 KB

  const int tid  = threadIdx.x;
  const int lane = tid & 31;
  const int wave = tid >> 5;
  const int m0   = (blockIdx.x * 4 + wave) * 16;   // 256 M-tiles total
  const int half = lane >> 4;                      // K-pair selector
  const int l16  = lane & 15;                      // M row (A) / N col (B)

  // Stage B tile: Bs[n][k] = (n < 10) ? Win[n][k] : 0
  for (int i = tid; i < 16 * K_IN; i += 128) {
    const int n = i >> 7;          // 0..15
    const int k = i & (K_IN - 1);  // 0..127
    Bs[n * BPAD + k] = (n < NQ) ? Win[n * K_IN + k] : 0.0f;
  }
  __syncthreads();

  const float* arow = x + (m0 + l16) * K_IN + half * 2;
  const float* brow = Bs + l16 * BPAD + half * 2;

  v8f acc = {};
  #pragma unroll 4
  for (int kk = 0; kk < K_IN / 4; ++kk) {
    v2f a, b;
    a.x = arow[kk * 4];
    a.y = arow[kk * 4 + 1];
    b.x = brow[kk * 4];
    b.y = brow[kk * 4 + 1];
    acc = __builtin_amdgcn_wmma_f32_16x16x4_f32(false, a, false, b,
                                                (short)0, acc, false, false);
  }
  if (l16 < NQ) {
    const float bias = bin[l16];
    #pragma unroll
    for (int r = 0; r < 8; ++r) {
      const int m = m0 + r + 8 * half;             // C/D layout: M = r + 8*half
      h[m * NQ + l16] = tanhf(acc[r] + bias);
    }
  }
}

// ---------------------------------------------------------------------------
// Kernel 2: 10-qubit VQC simulation. One wave32 per batch element; the
// 1024-amplitude complex state lives in LDS (8 KB/wave). 4 waves per block ->
// 34 KB LDS. RX gates are wave-private in-order DS read-modify-writes; the
// ring-CNOT permutation is staged through 64 VGPRs/lane. Z expectations are
// reduced with __shfl_xor (wave32).
// ---------------------------------------------------------------------------
__global__ __launch_bounds__(128) void k_circuit(
    const float* __restrict__ h, const float* __restrict__ qp,
    float* __restrict__ z)
{
  __shared__ float sre[4][DIM];
  __shared__ float sim[4][DIM];
  __shared__ unsigned short perm[DIM];

  const int tid  = threadIdx.x;
  const int wave = tid >> 5;
  const int lane = tid & 31;
  const int b    = blockIdx.x * 4 + wave;

  // Build ring-CNOT permutation table once per block:
  // new_state[j] = state[perm[j]], perm = p(0,1)∘p(1,2)∘...∘p(9,0) applied
  // innermost-first; p(c,t): flip target bit if control bit set.
  for (int j = tid; j < DIM; j += 128) {
    int src = j;
    if ((src >> 0) & 1) src ^= (1 << 9);               // (c=9, t=0)
    #pragma unroll
    for (int c = 8; c >= 0; --c)                       // (8,9) ... (0,1)
      if ((src >> (9 - c)) & 1) src ^= (1 << (9 - (c + 1)));
    perm[j] = (unsigned short)src;
  }
  __syncthreads();

  float* re = sre[wave];
  float* im = sim[wave];

  for (int k = 0; k < DIM / 32; ++k) {
    const int i = k * 32 + lane;
    re[i] = (i == 0) ? 1.0f : 0.0f;
    im[i] = 0.0f;
  }

  // RX(theta) on qubit w: pairs (i0, i0|mask), mask = 1<<(9-w).
  auto rx = [&](float theta, int w) {
    const float hc = cosf(0.5f * theta);
    const float hs = sinf(0.5f * theta);
    const int mask = 1 << (9 - w);
    const int low  = mask - 1;
    for (int k = 0; k < 16; ++k) {                     // 512 pairs / 32 lanes
      const int p  = k * 32 + lane;
      const int i0 = ((p & ~low) << 1) | (p & low);
      const int i1 = i0 | mask;
      const float are = re[i0], aim = im[i0];
      const float bre = re[i1], bim = im[i1];
      re[i0] = hc * are + hs * bim;                    // n0 = c*s0 - i*s*s1
      im[i0] = hc * aim - hs * bre;
      re[i1] = hc * bre + hs * aim;                    // n1 = -i*s*s0 + c*s1
      im[i1] = hc * bim - hs * are;
    }
  };

  for (int w = 0; w < NQ; ++w) rx(h[b * NQ + w], w);   // input encoding layer

  for (int l = 0; l < NL; ++l) {
    for (int w = 0; w < NQ; ++w) rx(qp[l * NQ + w], w);
    // ring-CNOT permutation: gather everything into registers, then write.
    float tre[32], tim[32];
    #pragma unroll
    for (int k = 0; k < 32; ++k) {
      const int s = perm[k * 32 + lane];
      tre[k] = re[s];
      tim[k] = im[s];
    }
    #pragma unroll
    for (int k = 0; k < 32; ++k) {
      const int j = k * 32 + lane;
      re[j] = tre[k];
      im[j] = tim[k];
    }
  }

  // z[b][q] = sum_i |amp_i|^2 * (1 - 2*bit_{9-q}(i))
  float acc[NQ];
  #pragma unroll
  for (int q = 0; q < NQ; ++q) acc[q] = 0.0f;
  for (int k = 0; k < 32; ++k) {
    const int i = k * 32 + lane;
    const float pr = re[i] * re[i] + im[i] * im[i];
    #pragma unroll
    for (int q = 0; q < NQ; ++q)
      acc[q] += ((i >> (9 - q)) & 1) ? -pr : pr;
  }
  #pragma unroll
  for (int q = 0; q < NQ; ++q) {
    float v = acc[q];
    for (int off = 16; off > 0; off >>= 1)
      v += __shfl_xor(v, off, 32);
    if (lane == 0) z[b * NQ + q] = v;
  }
}

// ---------------------------------------------------------------------------
// Kernel 3: out = z @ W_out^T + b_out   (4096x10)(10x256) -> (4096,256)
// One wave per 16x16 output tile; K=10 padded to 12 -> 3 WMMA f32 steps.
// Out-of-range K handled with clamped addresses + multiplicative masks on A
// (branch-free; 0 * finite-garbage == 0).
// ---------------------------------------------------------------------------
__global__ __launch_bounds__(128) void k_readout(
    const float* __restrict__ z, const float* __restrict__ Wout,
    const float* __restrict__ bout, float* __restrict__ out)
{
  const int lane = threadIdx.x & 31;
  const int wave = threadIdx.x >> 5;
  const int tile = blockIdx.x * 4 + wave;   // 256 M-tiles x 16 N-tiles = 4096
  const int m0   = (tile >> 4) * 16;
  const int n0   = (tile & 15) * 16;
  const int half = lane >> 4;
  const int l16  = lane & 15;

  const float* arow = z    + (m0 + l16) * NQ;
  const float* brow = Wout + (n0 + l16) * NQ;

  v8f acc = {};
  #pragma unroll
  for (int kk = 0; kk < 3; ++kk) {
    const int k0  = kk * 4 + half * 2;
    const int k0c = (k0 < NQ - 1) ? k0 : (NQ - 2);    // clamp: loads in-bounds
    const float vm0 = (k0     < NQ) ? 1.0f : 0.0f;
    const float vm1 = (k0 + 1 < NQ) ? 1.0f : 0.0f;
    v2f a, b;
    a.x = arow[k0c]     * vm0;
    a.y = arow[k0c + 1] * vm1;
    b.x = brow[k0c];
    b.y = brow[k0c + 1];
    acc = __builtin_amdgcn_wmma_f32_16x16x4_f32(false, a, false, b,
                                                (short)0, acc, false, false);
  }
  const float bias = bout[n0 + l16];
  #pragma unroll
  for (int r = 0; r < 8; ++r) {
    const int m = m0 + r + 8 * half;
    out[m * N_OUT + n0 + l16] = acc[r] + bias;
  }
}

// ---------------------------------------------------------------------------
extern "C" void kernel_launch(void* const* d_in, const int* in_sizes, int n_in,
                              void* d_out, int out_size, void* d_ws, size_t ws_size,
                              hipStream_t stream) {
  const float* x    = (const float*)d_in[0];   // (4096,128)
  const float* Win  = (const float*)d_in[1];   // (10,128)
  const float* bin  = (const float*)d_in[2];   // (10)
  const float* qp   = (const float*)d_in[3];   // (4,10)
  const float* Wout = (const float*)d_in[4];   // (256,10)
  const float* bout = (const float*)d_in[5];   // (256)
  float* out = (float*)d_out;                  // (4096,256) f32

  float* h  = (float*)d_ws;                    // (4096,10)
  float* zv = h + BATCH * NQ;                  // (4096,10)

  k_encode <<<BATCH / 16 / 4, 128, 0, stream>>>(x, Win, bin, h);
  k_circuit<<<BATCH / 4,      128, 0, stream>>>(h, qp, zv);
  k_readout<<<BATCH * 16 / 16 / 4, 128, 0, stream>>>(zv, Wout, bout, out);
}